// Attention_32779190403497
// MI455X (gfx1250) — compile-verified
//
#include <hip/hip_runtime.h>
#include <stdint.h>

// Problem constants (match reference).
#define B_ 64
#define T_ 2048
#define D_ 512

typedef __attribute__((ext_vector_type(2))) float v2f;
typedef __attribute__((ext_vector_type(8))) float v8f;

// ---------------------------------------------------------------------------
// Kernel 1: energy[b,t] = <K[b,t,:], q[b,:]>, apply mask (-inf), store to the
// attention region of d_out (raw masked logits; kernel 2 softmaxes in place).
//
// WMMA mapping (V_WMMA_F32_16X16X4_F32, C = A*B + C):
//   C[i,j] = sum_k A[i,k] * B[k,j]
//   A[i,k] = q[d0+k]        (replicated over i -> all C rows identical)
//   B[k,j] = K[t0+j, d0+k]  (16 t-values in N, 4 d-values in K)
// => C[*,j] accumulates energy[t0+j] over the d-loop.
//
// Per-lane operand layout (ISA 7.12.2, 32-bit A 16x4; B assumed mirrored):
//   lane L<16 : (v0,v1) hold K-dim {0,1};  lane L+16 : K-dim {2,3}
// so each lane feeds one float2 (8B, aligned) per operand per WMMA.
// ---------------------------------------------------------------------------
__global__ __launch_bounds__(256) void attn_energy_kernel(
    const float* __restrict__ q, const float* __restrict__ K,
    const uint8_t* __restrict__ mask, float* __restrict__ eout)
{
    __shared__ float qs[D_];                       // 2 KB of the 320 KB LDS
    const int b   = blockIdx.y;
    const int tid = threadIdx.x;

    // Stage q[b,:] in LDS (256 threads x float2).
    ((float2*)qs)[tid] = ((const float2*)(q + (size_t)b * D_))[tid];
    __syncthreads();

    const int lane = tid & 31;
    const int wave = tid >> 5;
    const int hi   = lane >> 4;    // 0: K-dim {0,1}, 1: K-dim {2,3}
    const int lj   = lane & 15;    // N index (t within tile)
    const float* Kb = K + (size_t)b * T_ * D_;

    #pragma unroll
    for (int tt = 0; tt < 2; ++tt) {
        const int t0 = blockIdx.x * 256 + wave * 32 + tt * 16;
        const float* Krow = Kb + (size_t)(t0 + lj) * D_;

        v8f c = {};
        #pragma unroll 4
        for (int d0 = 0; d0 < D_; d0 += 4) {
            v2f a = *(const v2f*)(qs   + d0 + 2 * hi);   // ds_load_b64
            v2f bb = *(const v2f*)(Krow + d0 + 2 * hi);  // global_load_b64
            c = __builtin_amdgcn_wmma_f32_16x16x4_f32(
                    false, a, false, bb, (short)0, c, false, false);
        }
        // All C rows identical; row 0 lives in c[0], N = lane for lanes 0..15.
        if (lane < 16) {
            const int t = t0 + lane;
            float e = c[0];
            if (mask[(size_t)b * T_ + t]) e = -__builtin_inff();
            eout[(size_t)b * T_ + t] = e;
        }
    }
}

// ---------------------------------------------------------------------------
// Kernel 2: in-place masked softmax over each row attn[b, 0..T).
// One workgroup (256 threads, 8 values/thread) owns one row -> in-place safe.
// ---------------------------------------------------------------------------
__global__ __launch_bounds__(256) void attn_softmax_kernel(float* __restrict__ attn)
{
    __shared__ float red[256];
    const int b   = blockIdx.x;
    const int tid = threadIdx.x;
    float* row = attn + (size_t)b * T_;

    float ev[8];
    float m = -__builtin_inff();
    #pragma unroll
    for (int k = 0; k < 8; ++k) { ev[k] = row[tid + 256 * k]; m = fmaxf(m, ev[k]); }

    red[tid] = m; __syncthreads();
    for (int s = 128; s > 0; s >>= 1) {
        if (tid < s) red[tid] = fmaxf(red[tid], red[tid + s]);
        __syncthreads();
    }
    m = red[0]; __syncthreads();

    float sum = 0.f;
    #pragma unroll
    for (int k = 0; k < 8; ++k) { ev[k] = __expf(ev[k] - m); sum += ev[k]; }

    red[tid] = sum; __syncthreads();
    for (int s = 128; s > 0; s >>= 1) {
        if (tid < s) red[tid] += red[tid + s];
        __syncthreads();
    }
    const float inv = 1.f / red[0];
    #pragma unroll
    for (int k = 0; k < 8; ++k) row[tid + 256 * k] = ev[k] * inv;
}

// ---------------------------------------------------------------------------
// Kernel 3: context[b,d] = sum_t attn[b,t] * V[b,t,d].
// WMMA mapping: A[i,k] = attn[t0+k] (replicated rows), B[k,j] = V[t0+k, d0+j].
// Each wave owns a 16-wide d-slice and accumulates 512 WMMAs over T.
// ---------------------------------------------------------------------------
__global__ __launch_bounds__(256) void attn_context_kernel(
    const float* __restrict__ attn, const float* __restrict__ V,
    float* __restrict__ ctx)
{
    __shared__ float as[T_];                       // 8 KB attention row cache
    const int b   = blockIdx.y;
    const int tid = threadIdx.x;

    const float4* ag = (const float4*)(attn + (size_t)b * T_);
    #pragma unroll
    for (int k = 0; k < 2; ++k)
        ((float4*)as)[tid + 256 * k] = ag[tid + 256 * k];   // 512 x b128
    __syncthreads();

    const int lane = tid & 31;
    const int wave = tid >> 5;
    const int hi   = lane >> 4;
    const int lj   = lane & 15;
    const int d0   = blockIdx.x * 128 + wave * 16;
    const float* Vb = V + (size_t)b * T_ * D_ + d0 + lj;

    v8f c = {};
    #pragma unroll 4
    for (int t0 = 0; t0 < T_; t0 += 4) {
        v2f a = *(const v2f*)(as + t0 + 2 * hi);            // ds_load_b64
        v2f bb;
        bb.x = Vb[(size_t)(t0 + 2 * hi)     * D_];          // global_load_b32
        bb.y = Vb[(size_t)(t0 + 2 * hi + 1) * D_];          // global_load_b32
        c = __builtin_amdgcn_wmma_f32_16x16x4_f32(
                false, a, false, bb, (short)0, c, false, false);
    }
    if (lane < 16)
        ctx[(size_t)b * D_ + d0 + lane] = c[0];
}

// ---------------------------------------------------------------------------
// Launch: inputs in setup_inputs() order: query, key, value, mask.
// d_out = context (B*D floats) followed by attention (B*T floats).
// Raw masked energy is staged directly in the attention region (no d_ws use).
// ---------------------------------------------------------------------------
extern "C" void kernel_launch(void* const* d_in, const int* in_sizes, int n_in,
                              void* d_out, int out_size, void* d_ws, size_t ws_size,
                              hipStream_t stream)
{
    (void)in_sizes; (void)n_in; (void)out_size; (void)d_ws; (void)ws_size;
    const float*   q    = (const float*)d_in[0];
    const float*   K    = (const float*)d_in[1];
    const float*   V    = (const float*)d_in[2];
    const uint8_t* mask = (const uint8_t*)d_in[3];   // jnp.bool_ = 1 byte/elem

    float* ctx  = (float*)d_out;
    float* attn = (float*)d_out + (size_t)B_ * D_;

    attn_energy_kernel <<<dim3(T_ / 256, B_), 256, 0, stream>>>(q, K, mask, attn);
    attn_softmax_kernel<<<B_,                256, 0, stream>>>(attn);
    attn_context_kernel<<<dim3(D_ / 128, B_), 256, 0, stream>>>(attn, V, ctx);
}